// GaborLayer_25477746000485
// MI455X (gfx1250) — compile-verified
//
#include <hip/hip_runtime.h>

typedef __attribute__((ext_vector_type(2))) float v2f;
typedef __attribute__((ext_vector_type(8))) float v8f;

#define WAVES_PER_BLOCK 8
#define POINTS_PER_WAVE 16
#define OUTF 256

// Gabor layer: out[p][o] = sin(x[p].W[o] + b[o]) * exp(-0.5*gamma[o]*||x[p]-mu[o]||^2)
// The two K=3 dot products are done with V_WMMA_F32_16X16X4_F32 (K padded to 4).
__global__ __launch_bounds__(256, 2) void gabor_wmma_kernel(
    const float* __restrict__ x,      // (P, 3)
    const float* __restrict__ W,      // (256, 3)
    const float* __restrict__ bias,   // (256,)
    const float* __restrict__ mu,     // (256, 3)
    const float* __restrict__ gamma,  // (256,)
    float* __restrict__ out)          // (P, 256)
{
    __shared__ float x2s[WAVES_PER_BLOCK * POINTS_PER_WAVE];

    const int  lane = threadIdx.x & 31;
    const int  wave = threadIdx.x >> 5;
    const int  m    = lane & 15;       // point-in-tile (lo half) / column n (B,C frags)
    const bool hi   = lane >= 16;

    const long tile = (long)blockIdx.x * WAVES_PER_BLOCK + wave;
    const long base = tile * POINTS_PER_WAVE;   // first point of this wave's tile

    // ---- A fragment: 16x4 f32 (one x point per row, K 3->4 zero padded) ----
    // lanes 0-15: VGPR0=x[m][0], VGPR1=x[m][1]; lanes 16-31: VGPR0=x[m][2], VGPR1=0
    const float* xp = x + (base + m) * 3;
    const float a0 = xp[0];
    const float a1 = xp[1];
    const float a2 = xp[2];
    v2f A;
    A.x = hi ? a2 : a0;
    A.y = hi ? 0.0f : a1;

    // ||x||^2 per point, broadcast via LDS (16 floats per wave)
    if (!hi) x2s[wave * 16 + m] = a0 * a0 + a1 * a1 + a2 * a2;
    __syncthreads();

    float x2v[8];
#pragma unroll
    for (int v = 0; v < 8; ++v)
        x2v[v] = x2s[wave * 16 + v + (hi ? 8 : 0)];

#pragma unroll 4
    for (int ot = 0; ot < OUTF / 16; ++ot) {
        const int o = ot * 16 + m;   // this lane's output feature (column n of B/C)

        // ---- B fragments: 4x16 f32 (K rows striped across lane halves) ----
        const float w0 = W[o * 3 + 0], w1 = W[o * 3 + 1], w2 = W[o * 3 + 2];
        const float u0 = mu[o * 3 + 0], u1 = mu[o * 3 + 1], u2 = mu[o * 3 + 2];
        v2f Bw, Bm;
        Bw.x = hi ? w2 : w0;  Bw.y = hi ? 0.0f : w1;
        Bm.x = hi ? u2 : u0;  Bm.y = hi ? 0.0f : u1;

        v8f clin = {};
        v8f ccrs = {};
        // D = A x B + 0   (8 args: neg_a, A, neg_b, B, c_mod, C, reuse_a, reuse_b)
        clin = __builtin_amdgcn_wmma_f32_16x16x4_f32(
            false, A, false, Bw, (short)0, clin, false, false);
        ccrs = __builtin_amdgcn_wmma_f32_16x16x4_f32(
            false, A, false, Bm, (short)0, ccrs, false, false);

        const float bb  = bias[o];
        const float gg  = gamma[o];
        const float mu2 = u0 * u0 + u1 * u1 + u2 * u2;

        float* orow = out + base * OUTF + o;
#pragma unroll
        for (int v = 0; v < 8; ++v) {
            const int   mp  = v + (hi ? 8 : 0);          // point row in tile
            const float lin = clin[v] + bb;
            const float D   = x2v[v] - 2.0f * ccrs[v] + mu2;
            const float r   = __sinf(lin) * __expf(-0.5f * gg * D);
            // 268 MB streaming output: nontemporal store (TH=NT)
            __builtin_nontemporal_store(r, orow + (long)mp * OUTF);
        }
    }
}

extern "C" void kernel_launch(void* const* d_in, const int* in_sizes, int n_in,
                              void* d_out, int out_size, void* d_ws, size_t ws_size,
                              hipStream_t stream) {
    const float* x     = (const float*)d_in[0];
    const float* W     = (const float*)d_in[1];
    const float* bias  = (const float*)d_in[2];
    const float* mu    = (const float*)d_in[3];
    const float* gamma = (const float*)d_in[4];
    float* out = (float*)d_out;

    const int n_points = in_sizes[0] / 3;                     // 262144
    const int n_tiles  = (n_points + POINTS_PER_WAVE - 1) / POINTS_PER_WAVE;
    const int blocks   = (n_tiles + WAVES_PER_BLOCK - 1) / WAVES_PER_BLOCK;  // 2048

    gabor_wmma_kernel<<<blocks, WAVES_PER_BLOCK * 32, 0, stream>>>(
        x, W, bias, mu, gamma, out);
}